// InvariantPointAttention_37546604101792
// MI455X (gfx1250) — compile-verified
//
#include <hip/hip_runtime.h>
#include <math.h>

// ---------------- problem constants ----------------
constexpr int B_  = 2;
constexpr int N_  = 512;
constexpr int D_  = 384;
constexpr int BN  = B_ * N_;          // 1024 rows
constexpr int HSQK = 192;             // H*SQK
constexpr int HSV  = 192;             // H*SV
constexpr int HPQ3 = 144;             // H*PQ*3
constexpr int HPV3 = 288;             // H*PV*3
constexpr int NHP  = 48;              // H*PQ point pairs
constexpr float SCALE_ = 0.17677669529663687f;   // 1/sqrt(32)
constexpr float EPS_   = 1e-12f;

typedef float v2f __attribute__((ext_vector_type(2)));
typedef float v4f __attribute__((ext_vector_type(4)));
typedef float v8f __attribute__((ext_vector_type(8)));
typedef int   v4i __attribute__((ext_vector_type(4)));

// fp32 WMMA: D(16x16,f32) = A(16x4,f32) * B(4x16,f32) + C
#define WMMA_F32X4(a, b, c) \
  __builtin_amdgcn_wmma_f32_16x16x4_f32(false, (a), false, (b), (short)0, (c), false, false)

// ---- async global->LDS staging (CDNA5 ASYNCcnt path), with safe fallback ----
// Signature probe-confirmed (round-2 diagnostic): typed non-const pointers,
// param0 in the global AS, param1 in LDS AS, plus two immediates.
#if __has_builtin(__builtin_amdgcn_global_load_async_to_lds_b32)
#define HAVE_ASYNC_LDS 1
#define ASYNC_COPY_B32(dst, src)                                               \
  __builtin_amdgcn_global_load_async_to_lds_b32(                               \
      (__attribute__((address_space(1))) int*)(src),                           \
      (__attribute__((address_space(3))) int*)(dst), 0, 0)
#else
#define HAVE_ASYNC_LDS 0
#define ASYNC_COPY_B32(dst, src) (*(dst) = *(src))
#endif

__device__ __forceinline__ void wait_async_lds() {
#if HAVE_ASYNC_LDS
#if __has_builtin(__builtin_amdgcn_s_wait_asynccnt)
  __builtin_amdgcn_s_wait_asynccnt(0);
#else
  asm volatile("s_wait_asynccnt 0x0" ::: "memory");
#endif
#endif
}

// ---------------------------------------------------------------------------
// fp32 WMMA GEMM with LDS-staged weights:
//   C[M,Nout] = A[M,K] @ W[K,Nout] (+ bias)
// Block = 128 threads (4 waves). Block tile = 64 rows x 48 cols:
//   wave w computes rows [bm*64 + w*16 .. +16), cols [bn*48 .. +48)
//   (3 accumulators per wave reuse one A fragment).
// Weight slab W[kc:kc+32, col0:col0+48] is staged in LDS via async b32
// copies (ASYNCcnt), **k-pair interleaved**:
//   lb[(kr/2)*96 + col*2 + (kr&1)] = W[k0+kr][col0+col]
// so each WMMA B fragment {W[kr][c], W[kr+1][c]} is one aligned ds_load_b64.
// Requires: M % 64 == 0, (Nout/16) % 3 == 0, K % 32 == 0.  (All calls comply.)
// ---------------------------------------------------------------------------
constexpr int KC = 32;

__global__ void gemm_wmma_f32(const float* __restrict__ A, int lda,
                              const float* __restrict__ W, int ldw,
                              const float* __restrict__ bias,
                              float* __restrict__ C, int ldc,
                              int M, int Nout, int K) {
  __shared__ float lb[KC * 48];                 // 6 KB weight slab
  const int tid  = threadIdx.x;
  const int lane = tid & 31;
  const int wave = tid >> 5;
  const int ntn3 = (Nout >> 4) / 3;
  const int bm   = blockIdx.x / ntn3;
  const int bn   = blockIdx.x - bm * ntn3;
  const int l15  = lane & 15;
  const int kh   = lane >> 4;                   // K-half: {0,1} vs {2,3}
  const int arow = (bm * 4 + wave) * 16 + l15;
  const int col0 = bn * 48;

  const float* ap = A + (size_t)arow * lda + 2 * kh;     // per-lane A stream
  v8f acc0 = {}, acc1 = {}, acc2 = {};

  for (int k0 = 0; k0 < K; k0 += KC) {
    // ---- stage slab (pair-interleaved) into LDS: 1536 x b32, 12/thread ----
#pragma unroll
    for (int t = 0; t < 12; ++t) {
      const int idx = tid + t * 128;            // element of the 32x48 slab
      const int kk = idx / 48, c = idx - kk * 48;
      const float* src = W + (size_t)(k0 + kk) * ldw + col0 + c;
      float* dst = &lb[(kk >> 1) * 96 + c * 2 + (kk & 1)];
      ASYNC_COPY_B32(dst, src);
    }
    wait_async_lds();
    __syncthreads();

    if (k0 + KC < K)                            // prefetch next A chunk
      __builtin_prefetch(ap + k0 + KC, 0, 0);

    // ---- compute: 8 k-steps x 3 wmma; B fragments = single ds_load_b64 ----
#pragma unroll
    for (int k = 0; k < KC; k += 4) {
      v2f a;
      a[0] = ap[k0 + k];
      a[1] = ap[k0 + k + 1];
      const int kp = (k >> 1) + kh;             // (k + 2*kh) / 2
      const v2f* bp = (const v2f*)&lb[kp * 96 + l15 * 2];
      const v2f b0 = bp[0];                     // cols l15
      const v2f b1 = bp[16];                    // cols l15+16
      const v2f b2 = bp[32];                    // cols l15+32
      acc0 = WMMA_F32X4(a, b0, acc0);
      acc1 = WMMA_F32X4(a, b1, acc1);
      acc2 = WMMA_F32X4(a, b2, acc2);
    }
    __syncthreads();
  }

  // ---- store 16x48 tile (+ bias) ----
  const int tm = bm * 4 + wave;
  const float bv0 = bias ? bias[col0 + l15]      : 0.0f;
  const float bv1 = bias ? bias[col0 + 16 + l15] : 0.0f;
  const float bv2 = bias ? bias[col0 + 32 + l15] : 0.0f;
#pragma unroll
  for (int r = 0; r < 8; ++r) {
    const int gi = tm * 16 + kh * 8 + r;
    float* crow = C + (size_t)gi * ldc + col0;
    crow[l15]      = acc0[r] + bv0;
    crow[16 + l15] = acc1[r] + bv1;
    crow[32 + l15] = acc2[r] + bv2;
  }
}

// ---------------------------------------------------------------------------
// Per (token, h, p): squared norm and sqrt(norm^2 + eps) of the 3-vector.
// ---------------------------------------------------------------------------
__global__ void point_norms(const float* __restrict__ qp,
                            float* __restrict__ n2, float* __restrict__ nrm) {
  const int idx = blockIdx.x * blockDim.x + threadIdx.x;
  if (idx >= BN * NHP) return;
  const float* p = qp + (size_t)(idx / NHP) * HPQ3 + (idx % NHP) * 3;
  const float s = p[0] * p[0] + p[1] * p[1] + p[2] * p[2];
  n2[idx]  = s;
  nrm[idx] = sqrtf(s + EPS_);
}

// ---------------------------------------------------------------------------
// Fused attention logits: one wave per 16x16 (i,j) tile of [B,N,N].
//   scalar part: q_s[i,:192] . k_s[j,:192]  via 48 fp32 WMMA (K=4) steps
//   point part : per (h,p) 16x16 dot tile via K=3-padded fp32 WMMA, then
//                fused dist+cos elementwise in registers.
// ---------------------------------------------------------------------------
__global__ void ipa_logits(const float* __restrict__ qk,   // [BN,384] q|k
                           const float* __restrict__ qp,   // [BN,144]
                           const float* __restrict__ n2,   // [BN,48]
                           const float* __restrict__ nrm,  // [BN,48]
                           float* __restrict__ logits) {   // [B,N,N]
  const int lane = threadIdx.x & 31;
  const int wave = threadIdx.x >> 5;
  const int TJ = N_ / 16;                                  // 32
  const int tile = blockIdx.x * (blockDim.x >> 5) + wave;  // 0..2047
  if (tile >= B_ * TJ * TJ) return;
  const int b   = tile / (TJ * TJ);
  const int rem = tile - b * TJ * TJ;
  const int ti = rem / TJ, tj = rem - (rem / TJ) * TJ;
  const int l15 = lane & 15;
  const int kh  = lane >> 4;
  const int iA = b * N_ + ti * 16 + l15;   // A-side row this lane loads
  const int jB = b * N_ + tj * 16 + l15;   // B-side row == output column row

  // ---- scalar q.k over K=192 ----
  const float* qrow = qk + (size_t)iA * 384 + 2 * kh;        // q_s
  const float* krow = qk + (size_t)jB * 384 + HSQK + 2 * kh; // k_s
  v8f acc = {};
  for (int k0 = 0; k0 < HSQK; k0 += 4) {
    v2f a, bb;
    a[0]  = qrow[k0];
    a[1]  = qrow[k0 + 1];
    bb[0] = krow[k0];
    bb[1] = krow[k0 + 1];
    acc = WMMA_F32X4(a, bb, acc);
  }

  // ---- point dist + cos over 48 (h,p) pairs (K=3, padded to 4) ----
  const float* pI = qp + (size_t)iA * HPQ3;
  const float* pJ = qp + (size_t)jB * HPQ3;
  for (int t = 0; t < NHP; ++t) {
    v2f a, bb;
    a[0]  = pI[t * 3 + 2 * kh];
    a[1]  = kh ? 0.0f : pI[t * 3 + 1];   // K=3 -> pad 4th slot with 0
    bb[0] = pJ[t * 3 + 2 * kh];
    bb[1] = kh ? 0.0f : pJ[t * 3 + 1];
    v8f z = {};
    v8f d = WMMA_F32X4(a, bb, z);        // 16x16 dot tile for this (h,p)

    const float n2j = n2[(size_t)jB * NHP + t];
    const float nrj = nrm[(size_t)jB * NHP + t];
#pragma unroll
    for (int r = 0; r < 8; ++r) {
      const int gi = b * N_ + ti * 16 + kh * 8 + r;
      const float n2i = n2[(size_t)gi * NHP + t];
      const float nri = nrm[(size_t)gi * NHP + t];
      const float dot = d[r];
      const float d2  = n2i + n2j - 2.0f * dot;
      const float dist = sqrtf(fmaxf(d2, 0.0f) + EPS_);
      const float cs   = dot / (nri * nrj);
      acc[r] += dist + cs;
    }
  }

  // ---- write scaled logits ----
#pragma unroll
  for (int r = 0; r < 8; ++r) {
    const int gi = ti * 16 + kh * 8 + r;
    const int gj = tj * 16 + l15;
    logits[((size_t)b * N_ + gi) * N_ + gj] = acc[r] * SCALE_;
  }
}

// ---------------------------------------------------------------------------
// In-place row softmax, one wave32 per row of length N_=512.
// ---------------------------------------------------------------------------
__global__ void softmax_rows(float* __restrict__ x) {
  const int lane = threadIdx.x & 31;
  const int wave = threadIdx.x >> 5;
  const int row = blockIdx.x * (blockDim.x >> 5) + wave;
  if (row >= BN) return;
  float* p = x + (size_t)row * N_;

  float m = -INFINITY;
  for (int j = lane; j < N_; j += 32) m = fmaxf(m, p[j]);
#pragma unroll
  for (int s = 16; s > 0; s >>= 1) m = fmaxf(m, __shfl_xor(m, s, 32));

  float sum = 0.0f;
  for (int j = lane; j < N_; j += 32) sum += __expf(p[j] - m);
#pragma unroll
  for (int s = 16; s > 0; s >>= 1) sum += __shfl_xor(sum, s, 32);

  const float inv = 1.0f / sum;
  for (int j = lane; j < N_; j += 32) p[j] = __expf(p[j] - m) * inv;
}

// ---------------------------------------------------------------------------
// xyz_out = xyz + mean over 128 of point projection (cat[:,384:768] as [.,128,3])
// ---------------------------------------------------------------------------
__global__ void xyz_update(const float* __restrict__ xyz,
                           const float* __restrict__ cat,   // [BN,768]
                           float* __restrict__ out) {        // [BN,3]
  const int idx = blockIdx.x * blockDim.x + threadIdx.x;
  if (idx >= BN * 3) return;
  const int row = idx / 3, c = idx - row * 3;
  const float* p = cat + (size_t)row * 768 + 384 + c;
  float s = 0.0f;
  for (int k = 0; k < 128; ++k) s += p[3 * k];
  out[idx] = xyz[idx] + s * (1.0f / 128.0f);
}

// ---------------------------------------------------------------------------
extern "C" void kernel_launch(void* const* d_in, const int* in_sizes, int n_in,
                              void* d_out, int out_size, void* d_ws, size_t ws_size,
                              hipStream_t stream) {
  const float* hidden = (const float*)d_in[0];
  const float* xyz    = (const float*)d_in[1];
  const float* W_qk  = (const float*)d_in[2];  const float* b_qk  = (const float*)d_in[3];
  const float* W_vs  = (const float*)d_in[4];  const float* b_vs  = (const float*)d_in[5];
  const float* W_so  = (const float*)d_in[6];  const float* b_so  = (const float*)d_in[7];
  const float* W_pqk = (const float*)d_in[8];  const float* b_pqk = (const float*)d_in[9];
  const float* W_pv  = (const float*)d_in[10]; const float* b_pv  = (const float*)d_in[11];
  const float* W_po  = (const float*)d_in[12]; const float* b_po  = (const float*)d_in[13];
  const float* W_fp  = (const float*)d_in[14]; const float* b_fp  = (const float*)d_in[15];

  float* out = (float*)d_out;                  // [BN*384] result, then [BN*3] xyz
  float* ws  = (float*)d_ws;

  // workspace layout (floats)
  float* qk    = ws;                           // [BN,384]  q_s | k_s
  float* v_s   = qk    + (size_t)BN * 384;     // [BN,192]
  float* qpb   = v_s   + (size_t)BN * HSV;     // [BN,144]
  float* vp    = qpb   + (size_t)BN * HPQ3;    // [BN,288]
  float* n2    = vp    + (size_t)BN * HPV3;    // [BN,48]
  float* nrm   = n2    + (size_t)BN * NHP;     // [BN,48]
  float* attn  = nrm   + (size_t)BN * NHP;     // [B,N,N]
  float* s_res = attn  + (size_t)B_ * N_ * N_; // [BN,192]
  float* p_res = s_res + (size_t)BN * HSV;     // [BN,288]
  float* cat   = p_res + (size_t)BN * HPV3;    // [BN,768]  (scalar|point proj)

  auto gemm = [&](const float* A, int lda, const float* W, int ldw,
                  const float* bias, float* C, int ldc, int M, int Nout, int K) {
    const int blocks = (M / 64) * ((Nout / 16) / 3);   // 64x48 block tiles
    gemm_wmma_f32<<<blocks, 128, 0, stream>>>(A, lda, W, ldw, bias, C, ldc, M, Nout, K);
  };

  // input projections (hidden: [BN,384])
  gemm(hidden, D_, W_qk,  384,  b_qk,  qk,  384,  BN, 384,  D_);
  gemm(hidden, D_, W_vs,  HSV,  b_vs,  v_s, HSV,  BN, HSV,  D_);
  gemm(hidden, D_, W_pqk, HPQ3, b_pqk, qpb, HPQ3, BN, HPQ3, D_);
  gemm(hidden, D_, W_pv,  HPV3, b_pv,  vp,  HPV3, BN, HPV3, D_);

  // point norms
  point_norms<<<(BN * NHP + 255) / 256, 256, 0, stream>>>(qpb, n2, nrm);

  // fused scalar+point logits: 2 * 32 * 32 = 2048 tiles, 4 waves/block
  ipa_logits<<<512, 128, 0, stream>>>(qk, qpb, n2, nrm, attn);

  // softmax: 1024 rows, 4 waves/block
  softmax_rows<<<256, 128, 0, stream>>>(attn);

  // attn @ v_s, attn @ vp (per batch)
  for (int b = 0; b < B_; ++b) {
    gemm(attn + (size_t)b * N_ * N_, N_, v_s + (size_t)b * N_ * HSV,  HSV,
         nullptr, s_res + (size_t)b * N_ * HSV,  HSV,  N_, HSV,  N_);
    gemm(attn + (size_t)b * N_ * N_, N_, vp  + (size_t)b * N_ * HPV3, HPV3,
         nullptr, p_res + (size_t)b * N_ * HPV3, HPV3, N_, HPV3, N_);
  }

  // output projections, written into concat buffer [BN,768]
  gemm(s_res, HSV,  W_so, D_, b_so, cat,       768, BN, D_, HSV);
  gemm(p_res, HPV3, W_po, D_, b_po, cat + 384, 768, BN, D_, HPV3);

  // final projection straight into d_out
  gemm(cat, 768, W_fp, D_, b_fp, out, D_, BN, D_, 768);

  // xyz + mean(point projection)
  xyz_update<<<(BN * 3 + 255) / 256, 256, 0, stream>>>(xyz, cat, out + (size_t)BN * D_);
}